// KANN_31379031064675
// MI455X (gfx1250) — compile-verified
//
#include <hip/hip_runtime.h>
#include <math.h>

// ---------------- problem constants (from reference) ----------------
#define KN_WIDTH     64
#define KN_ORDER     3
#define KN_ELEMENTS  64
#define KN_NODES     (KN_ELEMENTS * KN_ORDER + 1)   // 193
#define KN_BLOCK     256
#define KN_WTAB      (KN_WIDTH * KN_NODES)          // 12352 floats = 49408 B per table

// ---------------- CDNA5 async global->LDS staging -------------------
#if defined(__has_builtin)
#  if __has_builtin(__builtin_amdgcn_global_load_async_to_lds_b128)
#    define KN_HAVE_ASYNC_LDS 1
#  endif
#  if __has_builtin(__builtin_amdgcn_s_wait_asynccnt)
#    define KN_HAVE_WAIT_ASYNC 1
#  endif
#endif

// Exact pointee type required by the builtin: int vector of 16 bytes.
typedef int kn_v4i __attribute__((vector_size(16)));
typedef __attribute__((address_space(1))) kn_v4i kn_gv4i;   // global
typedef __attribute__((address_space(3))) kn_v4i kn_lv4i;   // LDS

// Cubic Lagrange basis at nodes {-1, -1/3, 1/3, 1}.
// phi_j(x) = prod_{m!=j} (x - n_m) / (n_j - n_m); constant denominators folded.
__device__ __forceinline__ void kn_lagrange4(float x, float p[4]) {
    const float a = x + 1.0f;              // (x - n0)
    const float b = x + (1.0f / 3.0f);     // (x - n1)
    const float c = x - (1.0f / 3.0f);     // (x - n2)
    const float d = x - 1.0f;              // (x - n3)
    p[0] = b * c * d * (-9.0f  / 16.0f);
    p[1] = a * c * d * ( 27.0f / 16.0f);
    p[2] = a * b * d * (-27.0f / 16.0f);
    p[3] = a * b * c * (  9.0f / 16.0f);
}

__global__ __launch_bounds__(KN_BLOCK)
void kann_fused_kernel(const float* __restrict__ xin,
                       const float* __restrict__ w_inner,
                       const float* __restrict__ w_outer,
                       float* __restrict__ out,
                       int n) {
    __shared__ float s_wi[KN_WTAB];
    __shared__ float s_wo[KN_WTAB];

    const int tid = threadIdx.x;

    // ---- stage both weight tables into LDS (98.8 KB total) ----
#if defined(KN_HAVE_ASYNC_LDS)
    {
        const char* gi = (const char*)w_inner;
        const char* go = (const char*)w_outer;
        char* li = (char*)s_wi;
        char* lo = (char*)s_wo;
        // KN_WTAB * 4 bytes = 3088 x 16-byte chunks per table (exact).
        for (int t = tid; t < (KN_WTAB / 4); t += KN_BLOCK) {
            __builtin_amdgcn_global_load_async_to_lds_b128(
                (kn_gv4i*)(gi + 16 * t), (kn_lv4i*)(li + 16 * t), 0, 0);
            __builtin_amdgcn_global_load_async_to_lds_b128(
                (kn_gv4i*)(go + 16 * t), (kn_lv4i*)(lo + 16 * t), 0, 0);
        }
#if defined(KN_HAVE_WAIT_ASYNC)
        __builtin_amdgcn_s_wait_asynccnt(0);
#else
        asm volatile("s_wait_asynccnt 0x0" ::: "memory");
#endif
    }
#else
    for (int t = tid; t < KN_WTAB; t += KN_BLOCK) {
        s_wi[t] = w_inner[t];
        s_wo[t] = w_outer[t];
    }
#endif
    __syncthreads();

    const int i = blockIdx.x * KN_BLOCK + tid;
    if (i >= n) return;

    const float x = xin[i];

    // ---- layer 1: element/basis shared across all 64 widths ----
    const float xs1 = (float)(KN_NODES - 1) * x;                 // (N-1)*(x-0)/(1-0)
    float fe1 = floorf(xs1 * (1.0f / (float)KN_ORDER));
    fe1 = fminf(fmaxf(fe1, 0.0f), (float)(KN_ELEMENTS - 1));
    const int base1 = (int)fe1 * KN_ORDER;                       // multiple of 3, in [0,189]
    const float xt1 = (xs1 - ((float)base1 + 1.5f)) * (1.0f / 1.5f);
    float p1[4];
    kn_lagrange4(xt1, p1);

    float acc = 0.0f;

#pragma unroll 4
    for (int k = 0; k < KN_WIDTH; ++k) {
        // layer-1 contraction: gather 4 consecutive weights from LDS
        const float* wr = &s_wi[k * KN_NODES + base1];
        const float t1 = p1[0] * wr[0] + p1[1] * wr[1] + p1[2] * wr[2] + p1[3] * wr[3];

        // ---- layer 2: per-(i,k) element/basis ----
        const float xs2 = (float)(KN_NODES - 1) * t1;
        float fe2 = floorf(xs2 * (1.0f / (float)KN_ORDER));
        fe2 = fminf(fmaxf(fe2, 0.0f), (float)(KN_ELEMENTS - 1));
        const int base2 = (int)fe2 * KN_ORDER;
        const float xt2 = (xs2 - ((float)base2 + 1.5f)) * (1.0f / 1.5f);
        float p2[4];
        kn_lagrange4(xt2, p2);

        const float* wo = &s_wo[k * KN_NODES + base2];
        acc += p2[0] * wo[0] + p2[1] * wo[1] + p2[2] * wo[2] + p2[3] * wo[3];
    }

    out[i] = acc;   // width-sum
}

extern "C" void kernel_launch(void* const* d_in, const int* in_sizes, int n_in,
                              void* d_out, int out_size, void* d_ws, size_t ws_size,
                              hipStream_t stream) {
    (void)n_in; (void)d_ws; (void)ws_size; (void)out_size;
    const float* x       = (const float*)d_in[0];
    const float* w_inner = (const float*)d_in[1];
    const float* w_outer = (const float*)d_in[2];
    float*       out     = (float*)d_out;
    const int n = in_sizes[0];   // N_SAMPLES = 131072

    const int grid = (n + KN_BLOCK - 1) / KN_BLOCK;
    kann_fused_kernel<<<grid, KN_BLOCK, 0, stream>>>(x, w_inner, w_outer, out, n);
}